// ODEblockW_28922309771809
// MI455X (gfx1250) — compile-verified
//
#include <hip/hip_runtime.h>
#include <cstddef>
#include <utility>

// ---------------------------------------------------------------------------
// ETD1 integrator: z_{k+1} = e^{dtA} z_k e^{dtB} + dt*phi1(dtA) x0
// Heavy math = fp32 GEMMs on V_WMMA_F32_16X16X4_F32 (full-precision WMMA),
// staged through LDS with double-buffered GLOBAL_LOAD_ASYNC_TO_LDS transfers
// (ASYNCcnt-tracked), B slabs stored K-pair-interleaved so every WMMA fragment
// is one ds_load_b64. The reference's linalg.solve is replaced by the phi1
// Taylor series (exact to fp32 since ||dt*A|| ~ 0.075).
// ---------------------------------------------------------------------------

typedef __attribute__((ext_vector_type(2))) float v2f;
typedef __attribute__((ext_vector_type(8))) float v8f;

#define DT_STEP 0.1f
#define N_STEPS 9            // int(1.0 // 0.1) == 9 under Python float floor-div
#define TAYLOR_ORDER 10
// squarings = 4  ->  pre-scale by 1/16, square 4x

#define BM 128
#define BN 128
#define BK 16
#define LDA_S 20    // A slab row stride: 80B (16B multiple, conflict-free frags)
// B slab: K-pair interleaved, exactly BK*128 floats, no padding needed.

// ---- CDNA5 async global->LDS + counter helpers ----------------------------
__device__ __forceinline__ unsigned lds_addr(const void* p) {
  // generic LDS pointer: low 32 bits are the LDS byte offset (aperture layout)
  return (unsigned)(unsigned long long)p;
}

__device__ __forceinline__ void async_ld_b128(const void* gaddr, unsigned loff) {
  asm volatile("global_load_async_to_lds_b128 %0, %1, off"
               :: "v"(loff), "v"(gaddr) : "memory");
}
__device__ __forceinline__ void async_ld_b32(const void* gaddr, unsigned loff) {
  asm volatile("global_load_async_to_lds_b32 %0, %1, off"
               :: "v"(loff), "v"(gaddr) : "memory");
}

template <int N>
__device__ __forceinline__ void wait_asynccnt() {
#if __has_builtin(__builtin_amdgcn_s_wait_asynccnt)
  __builtin_amdgcn_s_wait_asynccnt(N);
#else
  asm volatile("s_wait_asynccnt %0" :: "i"(N) : "memory");
#endif
}

// ------- GEMM: D = sAB*(A@B) + sC*Cadd ; optional Acc += accS*D  -----------
__global__ __launch_bounds__(256) void gemm_wmma_f32(
    const float* __restrict__ A, const float* __restrict__ B,
    const float* __restrict__ Cadd, float* __restrict__ D,
    float* __restrict__ Acc,
    int M, int N, int K, float sAB, float sC, float accS)
{
  __shared__ __align__(16) float As[2][BM * LDA_S];
  __shared__ __align__(16) float Bs[2][BK * 128];

  const int tid   = threadIdx.x;
  const int wave  = tid >> 5;
  const int lane  = tid & 31;
  const int wm    = (wave & 3) * 32;   // wave row offset in block tile
  const int wn    = (wave >> 2) * 64;  // wave col offset in block tile
  const int bM    = blockIdx.y * BM;
  const int bN    = blockIdx.x * BN;
  const int lhalf = lane >> 4;         // 0: K=+0,+1 / M=0..7 ; 1: K=+2,+3 / M=8..15
  const int lmod  = lane & 15;

  v8f acc[2][4];
#pragma unroll
  for (int i = 0; i < 2; ++i)
#pragma unroll
    for (int j = 0; j < 4; ++j)
#pragma unroll
      for (int e = 0; e < 8; ++e) acc[i][j][e] = 0.0f;

  // issue one K-slab (A: 2x b128/lane ; B: 8x b32/lane scattered into the
  // K-pair-interleaved layout) -> 10 async ops per lane per slab
  auto issue_slab = [&](int k0, int buf) {
    const int r0 = tid >> 2;             // 0..63
    const int c4 = (tid & 3) << 2;       // 0,4,8,12
#pragma unroll
    for (int p = 0; p < 2; ++p) {
      const int row = r0 + p * 64;
      async_ld_b128(A + (size_t)(bM + row) * K + k0 + c4,
                    lds_addr(&As[buf][row * LDA_S + c4]));
    }
#pragma unroll
    for (int p = 0; p < 8; ++p) {
      const int e   = p * 256 + tid;     // 0..2047
      const int k   = e >> 7;            // 0..15
      const int col = e & 127;
      const int off = ((k >> 2) * 2 + ((k >> 1) & 1)) * 256 + (col << 1) + (k & 1);
      async_ld_b32(B + (size_t)(k0 + k) * N + bN + col,
                   lds_addr(&Bs[buf][off]));
    }
  };

  const int nslab = K / BK;
  issue_slab(0, 0);

  for (int i = 0; i < nslab; ++i) {
    const int buf = i & 1;
    if (i + 1 < nslab) {
      issue_slab((i + 1) * BK, (i + 1) & 1);
      wait_asynccnt<10>();   // retire slab i (in-order), slab i+1 in flight
    } else {
      wait_asynccnt<0>();
    }
    __syncthreads();         // slab i visible to all waves

    const float* Asb = &As[buf][0];
    const float* Bsb = &Bs[buf][0];
#pragma unroll
    for (int kk = 0; kk < BK; kk += 4) {
      v2f af[2], bf[4];
#pragma unroll
      for (int ii = 0; ii < 2; ++ii)   // A 16x4 frag: contiguous K pair -> b64
        af[ii] = *reinterpret_cast<const v2f*>(
            &Asb[(wm + ii * 16 + lmod) * LDA_S + kk + 2 * lhalf]);
      const int bb = ((kk >> 2) * 2 + lhalf) * 256;
#pragma unroll
      for (int j = 0; j < 4; ++j)      // B 4x16 frag: interleaved pair -> b64
        bf[j] = *reinterpret_cast<const v2f*>(
            &Bsb[bb + ((wn + j * 16 + lmod) << 1)]);
#pragma unroll
      for (int ii = 0; ii < 2; ++ii)
#pragma unroll
        for (int j = 0; j < 4; ++j)
          acc[ii][j] = __builtin_amdgcn_wmma_f32_16x16x4_f32(
              false, af[ii], false, bf[j], (short)0, acc[ii][j], false, false);
    }
    __syncthreads();         // all waves done reading buf before it is reloaded
  }

  // ---- epilogue: C/D layout -> VGPR e, lane<16: M=e ; lane>=16: M=8+e ----
#pragma unroll
  for (int i = 0; i < 2; ++i) {
#pragma unroll
    for (int j = 0; j < 4; ++j) {
      const int col = bN + wn + j * 16 + lmod;
#pragma unroll
      for (int e = 0; e < 8; ++e) {
        const int row = bM + wm + i * 16 + lhalf * 8 + e;
        float v = sAB * acc[i][j][e];
        if (Cadd) v += sC * Cadd[(size_t)row * N + col];
        D[(size_t)row * N + col] = v;
        if (Acc) Acc[(size_t)row * N + col] += accS * v;  // fused series acc
      }
    }
  }
}

// ---------------- elementwise helpers ----------------
__global__ void k_scale(const float* __restrict__ X, float* __restrict__ Y,
                        float s, long n) {
  long i = (long)blockIdx.x * blockDim.x + threadIdx.x;
  if (i < n) Y[i] = s * X[i];
}

__global__ void k_identity(float* __restrict__ X, int n, float dval) {
  long i = (long)blockIdx.x * blockDim.x + threadIdx.x;
  long nn = (long)n * n;
  if (i < nn) {
    int r = (int)(i / n), c = (int)(i % n);
    X[i] = (r == c) ? dval : 0.0f;
  }
}

__global__ void k_subdiag(float* __restrict__ X, int n, float v) {
  int i = blockIdx.x * blockDim.x + threadIdx.x;
  if (i < n) X[(long)i * n + i] -= v;
}

// Z = dt * ( 0.5*sigmoid(alpha_r) * (adj - I) )
__global__ void k_build_A(const float* __restrict__ adj,
                          const float* __restrict__ alpha,
                          float* __restrict__ Z, int n, float dt) {
  long i = (long)blockIdx.x * blockDim.x + threadIdx.x;
  long nn = (long)n * n;
  if (i < nn) {
    int r = (int)(i / n), c = (int)(i % n);
    float sig = 1.0f / (1.0f + expf(-alpha[r]));
    float av = adj[i] - ((r == c) ? 1.0f : 0.0f);
    Z[i] = dt * 0.5f * sig * av;
  }
}

// o[i][j] = w[i][j] * clip(d[j], 0, 1)
__global__ void k_build_wdc(const float* __restrict__ w,
                            const float* __restrict__ d,
                            float* __restrict__ o, int h) {
  long i = (long)blockIdx.x * blockDim.x + threadIdx.x;
  long hh = (long)h * h;
  if (i < hh) {
    int c = (int)(i % h);
    float dc = fminf(fmaxf(d[c], 0.0f), 1.0f);
    o[i] = w[i] * dc;
  }
}

__global__ void k_transpose(const float* __restrict__ X, float* __restrict__ XT,
                            int rows, int cols) {
  __shared__ float tile[32][33];
  int x = blockIdx.x * 32 + threadIdx.x;
  int y = blockIdx.y * 32 + threadIdx.y;
#pragma unroll
  for (int j = 0; j < 32; j += 8)
    tile[threadIdx.y + j][threadIdx.x] = X[(size_t)(y + j) * cols + x];
  __syncthreads();
  int tx = blockIdx.y * 32 + threadIdx.x;
  int ty = blockIdx.x * 32 + threadIdx.y;
#pragma unroll
  for (int j = 0; j < 32; j += 8)
    XT[(size_t)(ty + j) * rows + tx] = tile[threadIdx.x][threadIdx.y + j];
}

// ---------------------------------------------------------------------------
extern "C" void kernel_launch(void* const* d_in, const int* in_sizes, int n_in,
                              void* d_out, int out_size, void* d_ws,
                              size_t ws_size, hipStream_t stream) {
  (void)in_sizes; (void)n_in; (void)out_size; (void)ws_size;

  const int N = 2048, H = 1024;
  const float* x     = (const float*)d_in[0];
  const float* x0    = (const float*)d_in[1];
  const float* adj   = (const float*)d_in[2];
  const float* alpha = (const float*)d_in[3];
  const float* w     = (const float*)d_in[4];
  const float* dvec  = (const float*)d_in[5];

  const size_t NNs = (size_t)N * N;
  const size_t HHs = (size_t)H * H;
  const size_t NHs = (size_t)N * H;

  float* ws  = (float*)d_ws;          // ~128 MB arena, resident in 192MB L2
  float* zA  = ws;                    // dt*A
  float* m1L = zA  + NNs;
  float* t0  = m1L + NNs;             // Taylor ping
  float* t1  = t0  + NNs;             // Taylor pong
  float* a0  = t1  + NNs;             // series accumulator / squaring ping
  float* a1  = a0  + NNs;             // squaring pong
  float* xs  = a1  + NNs;             // scaled matrix for expm
  float* m1R = xs  + NNs;
  float* zB  = m1R + HHs;
  float* wdc = zB  + HHs;
  float* wT  = wdc + HHs;
  // buffer reuse after the propagators are built:
  float* m2      = a0;   // phi1 accumulator (a0 free after expm(A))
  float* forcing = a1;   // N x H          (a1 free after expm(A))
  float* zbuf0   = xs;   // scan ping      (xs free after both expms)
  float* zbuf1   = t0;   // scan pong
  float* tmpNH   = t1;   // z @ m1_R intermediate

  auto blocks1d = [](long n) { return dim3((unsigned)((n + 255) / 256)); };

  auto gemm = [&](const float* Am, const float* Bm, const float* Cm, float* Dm,
                  float* Accm, int M, int Nn, int Kk, float sAB, float sC,
                  float accS) {
    dim3 grid(Nn / BN, M / BM), block(256);
    gemm_wmma_f32<<<grid, block, 0, stream>>>(Am, Bm, Cm, Dm, Accm,
                                              M, Nn, Kk, sAB, sC, accS);
  };
  auto scale = [&](const float* X, float* Y, float s, long n) {
    k_scale<<<blocks1d(n), 256, 0, stream>>>(X, Y, s, n);
  };
  auto setI = [&](float* X, int n, float dval) {
    k_identity<<<blocks1d((long)n * n), 256, 0, stream>>>(X, n, dval);
  };

  // expm(Z) by order-10 Taylor + 4 squarings (matches reference _expm);
  // the series accumulation a0 += Z^k/k! is fused into each GEMM epilogue.
  auto expm = [&](const float* Z, float* dest, int n) {
    const long nn = (long)n * n;
    scale(Z, xs, 1.0f / 16.0f, nn);      // 2^-squarings
    setI(t0, n, 1.0f);
    setI(a0, n, 1.0f);
    float* tp = t0; float* tn = t1;
    for (int k = 1; k <= TAYLOR_ORDER; ++k) {
      gemm(tp, xs, nullptr, tn, a0, n, n, n, 1.0f / (float)k, 0.0f, 1.0f);
      std::swap(tp, tn);
    }
    gemm(a0, a0, nullptr, a1, nullptr, n, n, n, 1.0f, 0.0f, 0.0f);
    gemm(a1, a1, nullptr, a0, nullptr, n, n, n, 1.0f, 0.0f, 0.0f);
    gemm(a0, a0, nullptr, a1, nullptr, n, n, n, 1.0f, 0.0f, 0.0f);
    gemm(a1, a1, nullptr, dest, nullptr, n, n, n, 1.0f, 0.0f, 0.0f);
  };

  // ---- B side: zB = dt*((w*d_c) @ w^T - I) ; m1R = expm(zB) ----
  k_build_wdc<<<blocks1d((long)HHs), 256, 0, stream>>>(w, dvec, wdc, H);
  k_transpose<<<dim3(H / 32, H / 32), dim3(32, 8), 0, stream>>>(w, wT, H, H);
  gemm(wdc, wT, nullptr, zB, nullptr, H, H, H, DT_STEP, 0.0f, 0.0f);
  k_subdiag<<<blocks1d(H), 256, 0, stream>>>(zB, H, DT_STEP);
  expm(zB, m1R, H);

  // ---- A side: zA = dt*A ; m1L = expm(zA) ----
  k_build_A<<<blocks1d((long)NNs), 256, 0, stream>>>(adj, alpha, zA, N, DT_STEP);
  expm(zA, m1L, N);

  // ---- m2 = A^{-1}(e^{dtA}-I) = dt * phi1(dt*A), phi1(Z)=sum Z^k/(k+1)! ----
  setI(t0, N, 1.0f);
  setI(m2, N, DT_STEP);                 // k=0 term: dt/1!
  {
    float* tp = t0; float* tn = t1;
    for (int k = 1; k <= TAYLOR_ORDER; ++k) {
      gemm(tp, zA, nullptr, tn, m2, N, N, N, 1.0f / (float)k, 0.0f,
           DT_STEP / (float)(k + 1));
      std::swap(tp, tn);
    }
  }
  gemm(m2, x0, nullptr, forcing, nullptr, N, H, N, 1.0f, 0.0f, 0.0f);

  // ---- scan: z <- m1L @ z @ m1R + forcing, 9 steps; last step -> d_out ----
  hipMemcpyAsync(zbuf0, x, NHs * sizeof(float), hipMemcpyDeviceToDevice, stream);
  float* zc = zbuf0; float* zn = zbuf1;
  for (int s = 0; s < N_STEPS; ++s) {
    gemm(zc, m1R, nullptr, tmpNH, nullptr, N, H, H, 1.0f, 0.0f, 0.0f);
    float* dst = (s == N_STEPS - 1) ? (float*)d_out : zn;
    gemm(m1L, tmpNH, forcing, dst, nullptr, N, H, N, 1.0f, 1.0f, 0.0f);
    std::swap(zc, zn);
  }
}